// StatPoolBackbone_9689446220478
// MI455X (gfx1250) — compile-verified
//
#include <hip/hip_runtime.h>
#include <hip/hip_bf16.h>
#include <math.h>
#include <stdint.h>

// Problem dims (fixed by the reference).
#define NB 64
#define NC 80
#define NT 16384
#define NTHREADS 256
#define EPSF 1e-6f

// gfx1250 async global->LDS copy (ASYNCcnt path). Guarded so the host pass
// and older toolchains still compile (fallback = plain float4 LDS staging).
#if defined(__HIP_DEVICE_COMPILE__) && \
    __has_builtin(__builtin_amdgcn_global_load_async_to_lds_b128)
#define HAVE_ASYNC_COPY 1
#else
#define HAVE_ASYNC_COPY 0
#endif

#if HAVE_ASYNC_COPY
// Builtin signature (from hipcc diagnostic): (global v4i*, local v4i*, Ii, Ii)
typedef int v4i __attribute__((ext_vector_type(4)));
typedef __attribute__((address_space(1))) v4i gv4i;
typedef __attribute__((address_space(3))) v4i lv4i;
#endif

__device__ __forceinline__ unsigned key_of(float f) {
  // Monotonic (totally ordered) unsigned key for IEEE-754 floats.
  unsigned u = __float_as_uint(f);
  return (u & 0x80000000u) ? ~u : (u | 0x80000000u);
}
__device__ __forceinline__ float float_of_key(unsigned k) {
  unsigned u = (k & 0x80000000u) ? (k & 0x7FFFFFFFu) : ~k;
  return __uint_as_float(u);
}

// Dynamic LDS layout: row[NT] | hist[6*256] | red[NTHREADS] | spfx[6] | skrm[6] | sbc[2]
#define SMEM_BYTES (NT * 4 + 6 * 256 * 4 + NTHREADS * 4 + 6 * 4 + 6 * 4 + 2 * 4)

__global__ __launch_bounds__(NTHREADS)
void statpool_kernel(const float* __restrict__ x,
                     const int* __restrict__ lengths,
                     float* __restrict__ out) {
  extern __shared__ unsigned char smem[];
  float*    row  = (float*)smem;                                        // NT floats
  unsigned* hist = (unsigned*)(smem + (size_t)NT * 4);                  // 6*256
  float*    red  = (float*)(smem + (size_t)NT * 4 + 6 * 256 * 4);      // NTHREADS
  unsigned* spfx = (unsigned*)(red + NTHREADS);                         // 6
  unsigned* skrm = spfx + 6;                                            // 6
  float*    sbc  = (float*)(skrm + 6);                                  // mean, std

  const int wg  = blockIdx.x;           // wg = b*NC + c
  const int b   = wg / NC;
  const int tid = threadIdx.x;
  const int n   = lengths[b];           // n in [NT/2, NT]
  const float* __restrict__ src = x + (size_t)wg * NT;

  // ---- Stage the whole row into LDS (single HBM read of this row) ----
#if HAVE_ASYNC_COPY
  #pragma unroll
  for (int i = 0; i < NT / (NTHREADS * 4); ++i) {
    const int e = (i * NTHREADS + tid) * 4;   // float index, 16B granules
    __builtin_amdgcn_global_load_async_to_lds_b128(
        (gv4i*)(src + e), (lv4i*)(row + e), 0, 0);
  }
#if __has_builtin(__builtin_amdgcn_s_wait_asynccnt)
  __builtin_amdgcn_s_wait_asynccnt(0);
#else
  asm volatile("s_wait_asynccnt 0" ::: "memory");
#endif
#else
  #pragma unroll
  for (int i = 0; i < NT / (NTHREADS * 4); ++i) {
    const int e = (i * NTHREADS + tid) * 4;
    const float4 v = *reinterpret_cast<const float4*>(src + e);
    *reinterpret_cast<float4*>(row + e) = v;
  }
#endif
  __syncthreads();

  // ---- Mean over valid prefix ----
  float acc = 0.0f;
  for (int i = tid; i < n; i += NTHREADS) acc += row[i];
  red[tid] = acc;
  __syncthreads();
  for (int s = NTHREADS >> 1; s > 0; s >>= 1) {
    if (tid < s) red[tid] += red[tid + s];
    __syncthreads();
  }
  if (tid == 0) sbc[0] = red[0] / (float)n;
  __syncthreads();
  const float mean = sbc[0];

  // ---- Variance (two-pass, matches reference) -> std ----
  float acc2 = 0.0f;
  for (int i = tid; i < n; i += NTHREADS) {
    const float d = row[i] - mean;
    acc2 += d * d;
  }
  red[tid] = acc2;
  __syncthreads();
  for (int s = NTHREADS >> 1; s > 0; s >>= 1) {
    if (tid < s) red[tid] += red[tid + s];
    __syncthreads();
  }
  if (tid == 0) sbc[1] = sqrtf(fmaxf(red[0] / (float)n, EPSF));

  // ---- Quantile ranks (same fp32 arithmetic as reference) ----
  const float n1 = (float)(n - 1);
  const float Q[3] = {0.25f, 0.5f, 0.75f};
  int lo[3], hi[3];
  float wq[3];
  #pragma unroll
  for (int q = 0; q < 3; ++q) {
    const float pos = Q[q] * n1;
    const float fl  = floorf(pos);
    lo[q] = (int)fl;
    hi[q] = (int)ceilf(pos);
    wq[q] = pos - fl;
  }
  if (tid < 6) {
    const int q = tid >> 1;
    skrm[tid] = (unsigned)((tid & 1) ? hi[q] : lo[q]);
    spfx[tid] = 0u;
  }
  __syncthreads();

  // ---- Concurrent 6-rank radix select, 8 bits/pass, 4 passes over LDS ----
  for (int p = 0; p < 4; ++p) {
    const int shift = 24 - 8 * p;
    const unsigned maskHi = p ? (0xFFFFFFFFu << (shift + 8)) : 0u;

    // Uniform per-thread snapshot of prefixes + dedupe map.
    unsigned pref[6];
    int rep[6];
    #pragma unroll
    for (int j = 0; j < 6; ++j) pref[j] = spfx[j];
    #pragma unroll
    for (int j = 0; j < 6; ++j) {
      rep[j] = j;
      for (int j2 = 0; j2 < j; ++j2)
        if (pref[j2] == pref[j]) { rep[j] = j2; break; }
    }

    for (int i = tid; i < 6 * 256; i += NTHREADS) hist[i] = 0u;
    __syncthreads();

    // One scan: each element lands in at most one distinct-prefix histogram.
    for (int i = tid; i < n; i += NTHREADS) {
      const unsigned K   = key_of(row[i]);
      const unsigned hb  = K & maskHi;
      const unsigned byt = (K >> shift) & 255u;
      #pragma unroll
      for (int j = 0; j < 6; ++j) {
        if (rep[j] == j && hb == pref[j])
          atomicAdd(&hist[j * 256 + byt], 1u);
      }
    }
    __syncthreads();

    // Bucket walk: thread j resolves rank j's next byte.
    if (tid < 6) {
      const int j = tid;
      const unsigned* h = &hist[rep[j] * 256];
      unsigned k = skrm[j];
      unsigned cum = 0u, bsel = 255u;
      for (int bkt = 0; bkt < 256; ++bkt) {
        const unsigned cnt = h[bkt];
        if (k < cum + cnt) { bsel = (unsigned)bkt; k -= cum; break; }
        cum += cnt;
      }
      skrm[j] = k;
      spfx[j] = pref[j] | (bsel << shift);
    }
    __syncthreads();
  }

  // ---- Emit (B, 5C): [mean | std | q25 | q50 | q75] ----
  if (tid == 0) {
    const size_t base = (size_t)b * (5 * NC) + (size_t)(wg - b * NC);
    out[base + 0 * NC] = sbc[0];
    out[base + 1 * NC] = sbc[1];
    #pragma unroll
    for (int q = 0; q < 3; ++q) {
      const float xlo = float_of_key(spfx[2 * q]);
      const float xhi = float_of_key(spfx[2 * q + 1]);
      out[base + (2 + q) * NC] = xlo + wq[q] * (xhi - xlo);
    }
  }
}

extern "C" void kernel_launch(void* const* d_in, const int* in_sizes, int n_in,
                              void* d_out, int out_size, void* d_ws, size_t ws_size,
                              hipStream_t stream) {
  (void)in_sizes; (void)n_in; (void)d_ws; (void)ws_size; (void)out_size;
  const float* x       = (const float*)d_in[0];
  const int*   lengths = (const int*)d_in[1];
  float*       out     = (float*)d_out;

  // Allow >64KB dynamic LDS (row + histograms). Deterministic, not a stream op.
  (void)hipFuncSetAttribute((const void*)statpool_kernel,
                            hipFuncAttributeMaxDynamicSharedMemorySize,
                            (int)SMEM_BYTES);

  statpool_kernel<<<NB * NC, NTHREADS, SMEM_BYTES, stream>>>(x, lengths, out);
}